// DeepLinearNet_39316130627713
// MI455X (gfx1250) — compile-verified
//
#include <hip/hip_runtime.h>

typedef __attribute__((ext_vector_type(2))) float v2f;
typedef __attribute__((ext_vector_type(8))) float v8f;

// ---------------------------------------------------------------------------
// Kernel 1: collapse the 64-layer chain of 3x3 matrices into one 3x3 matrix.
// out = x @ W0^T @ ... @ W63^T == x @ (W63 @ ... @ W0)^T ;  M = W63..W0.
// fp64, single thread (~1.7K FMAs, sub-microsecond).
// ---------------------------------------------------------------------------
__global__ void collapse_chain_kernel(const float* __restrict__ W,
                                      float* __restrict__ Mout, int depth) {
  if (threadIdx.x == 0 && blockIdx.x == 0) {
    double M[9];
#pragma unroll
    for (int i = 0; i < 9; ++i) M[i] = (double)W[i];
    for (int l = 1; l < depth; ++l) {
      const float* Wl = W + (size_t)l * 9;
      double w[9];
#pragma unroll
      for (int i = 0; i < 9; ++i) w[i] = (double)Wl[i];
      double T[9];
#pragma unroll
      for (int r = 0; r < 3; ++r)
#pragma unroll
        for (int c = 0; c < 3; ++c)
          T[r * 3 + c] = w[r * 3 + 0] * M[0 + c] +
                         w[r * 3 + 1] * M[3 + c] +
                         w[r * 3 + 2] * M[6 + c];
#pragma unroll
      for (int i = 0; i < 9; ++i) M[i] = T[i];
    }
#pragma unroll
    for (int i = 0; i < 9; ++i) Mout[i] = (float)M[i];
  }
}

// ---------------------------------------------------------------------------
// Kernel 2: out[i] = M * x[i], streamed at HBM rate.
// Transposed WMMA formulation (V_WMMA_F32_16X16X4_F32):
//   D(16x16) = A(16x4) x B(4x16)
//   A[m][k] = M[m][k]        (m<3, k<3; zero-pad)   -> wave-invariant operand
//   B[k][n] = x[r0+n][k]     (k<3; K=3 zero-pad)    -> 16 rows per WMMA
//   D[m][n] = out[r0+n][m]
//
// CDNA5 VGPR layouts (CDNA5 ISA):
//   A 16x4 f32 : VGPR0 = (M=Lm, K = hi?2:0) ; VGPR1 = (M=Lm, K = hi?3:1)
//   B 4x16 f32 : VGPR0 = (K = hi?2:0, N=Lm) ; VGPR1 = (K = hi?1:3... K = hi?3:1, N=Lm)
//   D 16x16 f32: VGPR v, lane L -> (M = v + 8*(L>=16), N = L&15)
// Useful output = d0..d2 in lanes 0..15: lane n holds out[r0+n][0..2],
// i.e. 12 contiguous bytes per lane, 192 contiguous bytes per wave-tile.
// ---------------------------------------------------------------------------
__global__ void apply_kernel(const float* __restrict__ x,
                             const float* __restrict__ Mw,
                             float* __restrict__ out, int nrows) {
  const int lane = threadIdx.x & 31;
  const int Lm = lane & 15;
  const bool hi = lane >= 16;

  // A = M zero-padded to 16x4, built once per wave.
  v2f a;
  a.x = (Lm < 3) ? Mw[Lm * 3 + (hi ? 2 : 0)] : 0.0f;   // K = 0 (lo) / 2 (hi)
  a.y = (Lm < 3 && !hi) ? Mw[Lm * 3 + 1] : 0.0f;       // K = 1 (lo) / 3 = 0

  const int nFull = nrows >> 4;                 // full 16-row tiles
  const int wavesPerBlock = blockDim.x >> 5;
  const int waveStride = gridDim.x * wavesPerBlock;
  const int wid = blockIdx.x * wavesPerBlock + (threadIdx.x >> 5);

  // Streaming per-lane pointers, bumped by a constant stride per iteration.
  const float* p = x + (size_t)(wid * 16 + Lm) * 3;   // this lane's input row
  float* q = out + (size_t)(wid * 16 + Lm) * 3;       // used by lanes 0..15
  const size_t step = (size_t)waveStride * 48;        // 16 rows * 3 floats

  for (int t = wid; t < nFull; t += waveStride) {     // wave-uniform loop
    // B operand: lo lanes load (x0, x1); hi lanes load (x2, 0).
    v2f b;
    b.x = p[hi ? 2 : 0];                 // one load, selected address
    float t1 = p[hi ? 0 : 1];            // always-valid load (no exec branch)
    b.y = hi ? 0.0f : t1;                // cndmask
    v8f c = {};
    v8f d = __builtin_amdgcn_wmma_f32_16x16x4_f32(
        /*neg_a=*/false, a, /*neg_b=*/false, b,
        /*c_mod=*/(short)0, c, /*reuse_a=*/false, /*reuse_b=*/false);
    if (!hi) {                           // lanes 0..15: 12B contiguous each
      q[0] = d[0];
      q[1] = d[1];
      q[2] = d[2];
    }
    p += step;
    q += step;
  }

  // Cold tail: rows not covered by full tiles (none when nrows % 16 == 0).
  const int tailStart = nFull << 4;
  const int rem = nrows - tailStart;
  if (rem > 0 && blockIdx.x == 0 && (int)threadIdx.x < rem) {
    const int r = tailStart + threadIdx.x;
    const float x0 = x[r * 3 + 0], x1 = x[r * 3 + 1], x2 = x[r * 3 + 2];
#pragma unroll
    for (int m = 0; m < 3; ++m)
      out[r * 3 + m] = Mw[m * 3 + 0] * x0 + Mw[m * 3 + 1] * x1 + Mw[m * 3 + 2] * x2;
  }
}

extern "C" void kernel_launch(void* const* d_in, const int* in_sizes, int n_in,
                              void* d_out, int out_size, void* d_ws, size_t ws_size,
                              hipStream_t stream) {
  const float* x = (const float*)d_in[0];   // (BATCH, 3) fp32
  const float* W = (const float*)d_in[1];   // (DEPTH, 3, 3) fp32
  float* out = (float*)d_out;               // (BATCH, 3) fp32
  float* Mws = (float*)d_ws;                // 9 floats of scratch

  const int nrows = in_sizes[0] / 3;
  const int depth = in_sizes[1] / 9;

  // Step 1: collapse the chain (tiny, fp64).
  collapse_chain_kernel<<<1, 32, 0, stream>>>(W, Mws, depth);

  // Step 2: bandwidth-bound streaming apply. 200 MB @ 23.3 TB/s ~= 8.6 us.
  const int threads = 256;                          // 8 waves/block
  const int wavesPerBlock = threads / 32;
  const long long nTiles = (nrows + 15) / 16;
  long long blocksNeeded = (nTiles + wavesPerBlock - 1) / wavesPerBlock;
  int blocks = (blocksNeeded < 8192) ? (int)blocksNeeded : 8192;  // 64K waves
  if (blocks < 1) blocks = 1;
  apply_kernel<<<blocks, threads, 0, stream>>>(x, Mws, out, nrows);
}